// FCCaps_29231547417193
// MI455X (gfx1250) — compile-verified
//
#include <hip/hip_runtime.h>
#include <math.h>

// Problem constants (from reference): b=8, n_l=1024, n_h=16, d_l=16, d_h=16
#define NB   8
#define NL   1024
#define NH   16
#define DL   16
#define DH   16
#define EPSQ 1e-20f

typedef __attribute__((ext_vector_type(2))) float v2f;
typedef __attribute__((ext_vector_type(8))) float v8f;

// ---------------------------------------------------------------------------
// Kernel 1: U_hat[b,i,k,l] = sum_j U_l[b,i,j] * W[i,k,j,l]
// One wave32 per (i,k) pair: D(16x16) = A(16x4) x B(4x16) chained 4x over K=j.
// A: M = batch (rows 8..15 zero-padded via value-select, NOT predicated loads,
//    so no exec-mask branching in the WMMA chain).  B: K = j, N = l.
// Uses V_WMMA_F32_16X16X4_F32 (exact fp32 path).
// ---------------------------------------------------------------------------
__global__ void __launch_bounds__(256)
caps_uhat(const float* __restrict__ Ul,   // [NB][NL][DL]
          const float* __restrict__ W,    // [NL][NH][DL][DH]
          float* __restrict__ Uhat)       // [NB][NL][NH][DH]
{
    const int lane = threadIdx.x & 31;
    const int wave = blockIdx.x * (blockDim.x >> 5) + (threadIdx.x >> 5);
    const int i = wave >> 4;        // capsule index 0..1023
    const int k = wave & 15;        // high-caps index 0..15
    const int half = lane >> 4;     // 0: lanes 0-15, 1: lanes 16-31
    const int m = lane & 15;        // A-row (batch) / B-D column (l)

    // ---- A fragment: load unconditionally from a clamped row, zero by select.
    const bool valid = (m < NB);
    const int  arow  = valid ? m : 0;                 // safe address for pad lanes
    const float* __restrict__ arowp = Ul + (arow * NL + i) * DL;

    v2f a[4];
#pragma unroll
    for (int s = 0; s < 4; ++s) {
        // lanes 0-15 need j = 4s+0,4s+1 ; lanes 16-31 need j = 4s+2,4s+3
        const v2f t = *(const v2f*)(arowp + 4 * s + 2 * half);  // 8B-aligned b64 load
        a[s].x = valid ? t.x : 0.0f;                  // v_cndmask, no branches
        a[s].y = valid ? t.y : 0.0f;
    }

    // ---- B tile + WMMA chain (back-to-back, no divergence in between).
    const float* __restrict__ wik = W + ((i * NH + k) * DL) * DH; // W[i][k][.][.]
    v8f c = {};
#pragma unroll
    for (int s = 0; s < 4; ++s) {
        const int j0 = 4 * s + 2 * half;
        v2f bf;
        bf.x = wik[ j0      * DH + m];   // row j0 striped across lanes, col N=m
        bf.y = wik[(j0 + 1) * DH + m];
        // (neg_a, A, neg_b, B, c_mod, C, reuse_a, reuse_b)
        c = __builtin_amdgcn_wmma_f32_16x16x4_f32(false, a[s], false, bf,
                                                  (short)0, c, false, false);
    }

    // D layout: lanes 0-15, VGPR v -> element (M=v, N=lane). Rows 0..7 = batches.
    // Stores for fixed v are contiguous 64B across lanes 0-15.
    if (lane < 16) {
#pragma unroll
        for (int v = 0; v < NB; ++v) {
            Uhat[((v * NL + i) * NH + k) * DH + lane] = c[v];
        }
    }
}

// ---------------------------------------------------------------------------
// Kernel 2: S[b,k,l] = sum_i U_hat[b,i,k,l]   (2048 outputs, deterministic)
// ---------------------------------------------------------------------------
__global__ void __launch_bounds__(256)
reduce_S(const float* __restrict__ Uhat, float* __restrict__ S)
{
    const int t  = blockIdx.x * 256 + threadIdx.x;   // 0..2047
    const int b  = t >> 8;
    const int kl = t & 255;
    float acc = 0.0f;
    for (int i = 0; i < NL; ++i)
        acc += Uhat[(b * NL + i) * (NH * DH) + kl];
    S[t] = acc;
}

// ---------------------------------------------------------------------------
// Kernel 3: per (b,h): A_sum[k] = (1/sqrt(DL)) * sum_l U_hat[b,h,k,l]*S[b,k,l]
//           CB[k] = softmax_k(A_sum) + B[h,k]
//           block-partial: part[blk][k*16+l] = sum_{h in blk} U_hat[b,h,k,l]*CB[k]
// Grid: 32 blocks x 256 threads; each block covers 256 h of a single b.
// ---------------------------------------------------------------------------
__global__ void __launch_bounds__(256)
caps_route(const float* __restrict__ Uhat,
           const float* __restrict__ S,
           const float* __restrict__ Bmat,   // [NL][NH]
           float* __restrict__ part)         // [gridDim.x][NH*DH]
{
    __shared__ float sS[NH * DH];
    __shared__ float sAcc[NH * DH];

    const int t = blockIdx.x * 256 + threadIdx.x;    // 0..8191
    const int b = t >> 10;
    const int h = t & (NL - 1);

    sS[threadIdx.x]   = S[b * (NH * DH) + threadIdx.x];
    sAcc[threadIdx.x] = 0.0f;
    __syncthreads();

    const float* __restrict__ row = Uhat + (b * NL + h) * (NH * DH); // contiguous 1KB
    const float invScale = 0.25f;  // 1/sqrt(DL), DL=16

    float a[NH];
    float mx = -3.402823466e+38f;
#pragma unroll
    for (int k = 0; k < NH; ++k) {
        float d = 0.0f;
#pragma unroll
        for (int l = 0; l < DH; ++l)
            d += row[k * DH + l] * sS[k * DH + l];
        a[k] = d * invScale;
        mx = fmaxf(mx, a[k]);
    }
    float denom = 0.0f;
#pragma unroll
    for (int k = 0; k < NH; ++k) {
        a[k] = __expf(a[k] - mx);
        denom += a[k];
    }
    const float rden = 1.0f / denom;
#pragma unroll
    for (int k = 0; k < NH; ++k)
        a[k] = a[k] * rden + Bmat[h * NH + k];       // CB[b,h,k]

    // accumulate this h's contribution into block-local U_h partial (LDS atomics)
#pragma unroll
    for (int k = 0; k < NH; ++k) {
        const float cb = a[k];
#pragma unroll
        for (int l = 0; l < DH; ++l)
            atomicAdd(&sAcc[k * DH + l], row[k * DH + l] * cb);
    }
    __syncthreads();

    part[blockIdx.x * (NH * DH) + threadIdx.x] = sAcc[threadIdx.x];
}

// ---------------------------------------------------------------------------
// Kernel 4: sum the 4 block-partials per b, then squash over l.
// 128 threads: one per (b,k) row.
// ---------------------------------------------------------------------------
__global__ void __launch_bounds__(128)
caps_squash(const float* __restrict__ part,  // [32][NH*DH], blocks 4b..4b+3 belong to b
            float* __restrict__ out)         // [NB][NH][DH]
{
    const int t = threadIdx.x;   // 0..127
    const int b = t >> 4;
    const int k = t & 15;

    float v[DH];
    float ss = 0.0f;
#pragma unroll
    for (int l = 0; l < DH; ++l) {
        float acc = 0.0f;
#pragma unroll
        for (int blk = 0; blk < 4; ++blk)
            acc += part[(b * 4 + blk) * (NH * DH) + k * DH + l];
        v[l] = acc;
        ss += acc * acc;
    }
    const float nrm  = sqrtf(ss);
    const float coef = (1.0f - 1.0f / (__expf(nrm) + EPSQ)) / (nrm + EPSQ);
#pragma unroll
    for (int l = 0; l < DH; ++l)
        out[(b * NH + k) * DH + l] = coef * v[l];
}

// ---------------------------------------------------------------------------
extern "C" void kernel_launch(void* const* d_in, const int* in_sizes, int n_in,
                              void* d_out, int out_size, void* d_ws, size_t ws_size,
                              hipStream_t stream)
{
    const float* Ul   = (const float*)d_in[0];  // (8,1024,16)
    const float* W    = (const float*)d_in[1];  // (1024,16,16,16)
    const float* Bmat = (const float*)d_in[2];  // (1024,16)
    float* out = (float*)d_out;                 // (8,16,16)

    float* Uhat = (float*)d_ws;                       // 8*1024*256 floats = 8 MB
    float* S    = Uhat + (size_t)NB * NL * NH * DH;   // 2048 floats
    float* part = S + NB * NH * DH;                   // 32*256 floats

    // 16384 waves (one per (i,k)), 8 waves per 256-thread block
    caps_uhat  <<<(NL * NH) / 8, 256, 0, stream>>>(Ul, W, Uhat);
    reduce_S   <<<(NB * NH * DH) / 256, 256, 0, stream>>>(Uhat, S);
    caps_route <<<(NB * NL) / 256, 256, 0, stream>>>(Uhat, S, Bmat, part);
    caps_squash<<<1, 128, 0, stream>>>(part, out);
}